// MoELayer_27341761806561
// MI455X (gfx1250) — compile-verified
//
#include <hip/hip_runtime.h>
#include <math.h>

typedef float v2f __attribute__((ext_vector_type(2)));
typedef float v8f __attribute__((ext_vector_type(8)));

#define DIM     1024
#define FF      2048
#define HCOLS   4096     // 2*FF
#define NEXP    8
#define M_TILE  32
#define F_CHUNK 128
#define SX      1028     // padded LDS row stride for X tile (floats)
#define SA      132      // padded LDS row stride for act chunk (floats)

static __device__ __forceinline__ v8f wmma4(v2f a, v2f b, v8f c) {
  // D = A(16x4 f32) * B(4x16 f32) + C(16x16 f32)
  return __builtin_amdgcn_wmma_f32_16x16x4_f32(false, a, false, b, (short)0, c,
                                               false, false);
}

// Async global -> LDS copy of 16 bytes (CDNA5 GLOBAL_LOAD_ASYNC_TO_LDS_B128,
// tracked by ASYNCcnt). lds generic pointer's low 32 bits are the LDS byte
// address (ISA 10.2 aperture rules).
static __device__ __forceinline__ void async_copy_b128(void* lds_dst, const void* gsrc) {
#if defined(__AMDGCN__)
  unsigned lds_off = (unsigned)(unsigned long long)lds_dst;
  asm volatile("global_load_async_to_lds_b128 %0, %1, off"
               :: "v"(lds_off), "v"(gsrc) : "memory");
#else
  *(float4*)lds_dst = *(const float4*)gsrc;
#endif
}

static __device__ __forceinline__ void async_wait_all() {
#if defined(__AMDGCN__)
  asm volatile("s_wait_asynccnt 0x0" ::: "memory");
#endif
}

__global__ void zero_out_kernel(float4* __restrict__ out, int n4) {
  int i = blockIdx.x * blockDim.x + threadIdx.x;
  if (i < n4) out[i] = make_float4(0.f, 0.f, 0.f, 0.f);
}

__global__ void init_lists_kernel(int* __restrict__ counts, int* __restrict__ tok,
                                  float* __restrict__ wgt, int n) {
  int i = blockIdx.x * blockDim.x + threadIdx.x;
  if (i < NEXP) counts[i] = 0;
  if (i < n) { tok[i] = 0; wgt[i] = 0.f; }
}

// One wave (32 lanes) per token: 8 gate dots, softmax, top-2, scatter to lists.
__global__ void router_kernel(const float* __restrict__ x, const float* __restrict__ Wg,
                              const float* __restrict__ bg, int* __restrict__ counts,
                              int* __restrict__ tok, float* __restrict__ wgt, int T) {
  int wave = threadIdx.x >> 5;
  int lane = threadIdx.x & 31;
  int t = blockIdx.x * 8 + wave;
  if (t >= T) return;

  float acc[NEXP];
#pragma unroll
  for (int e = 0; e < NEXP; ++e) acc[e] = 0.f;

  const float* xrow = x + (size_t)t * DIM;
  for (int d = lane; d < DIM; d += 32) {
    float xv = xrow[d];
    const float4* wr = (const float4*)(Wg + (size_t)d * NEXP);
    float4 w0 = wr[0], w1 = wr[1];
    acc[0] += xv * w0.x; acc[1] += xv * w0.y; acc[2] += xv * w0.z; acc[3] += xv * w0.w;
    acc[4] += xv * w1.x; acc[5] += xv * w1.y; acc[6] += xv * w1.z; acc[7] += xv * w1.w;
  }
#pragma unroll
  for (int off = 16; off >= 1; off >>= 1) {
#pragma unroll
    for (int e = 0; e < NEXP; ++e) acc[e] += __shfl_xor(acc[e], off, 32);
  }

  if (lane == 0) {
    float lg[NEXP], p[NEXP];
    float mx = -3.0e38f;
#pragma unroll
    for (int e = 0; e < NEXP; ++e) { lg[e] = acc[e] + bg[e]; mx = fmaxf(mx, lg[e]); }
    float s = 0.f;
#pragma unroll
    for (int e = 0; e < NEXP; ++e) { p[e] = __expf(lg[e] - mx); s += p[e]; }
    float inv = 1.f / s;
#pragma unroll
    for (int e = 0; e < NEXP; ++e) p[e] *= inv;
    int i0 = 0;
#pragma unroll
    for (int e = 1; e < NEXP; ++e) if (p[e] > p[i0]) i0 = e;
    int i1 = (i0 == 0) ? 1 : 0;
#pragma unroll
    for (int e = 0; e < NEXP; ++e) if (e != i0 && p[e] > p[i1]) i1 = e;
    float denom = p[i0] + p[i1] + 1e-6f;
    float w0 = p[i0] / denom, w1 = p[i1] / denom;
    int pos0 = atomicAdd(&counts[i0], 1);
    tok[i0 * T + pos0] = t;  wgt[i0 * T + pos0] = w0;
    int pos1 = atomicAdd(&counts[i1], 1);
    tok[i1 * T + pos1] = t;  wgt[i1 * T + pos1] = w1;
  }
}

// Fused expert MLP: block = (expert e, tile of 32 routed tokens).
// Stage 1: h = X(32x1024) @ W1[e], SwiGLU fused in-registers per 128-col chunk.
// Stage 2: y(32x1024) += act_chunk @ W2[e], persistent v8f accumulators.
__launch_bounds__(256, 1)
__global__ void moe_mlp_kernel(const float* __restrict__ x,
                               const float* __restrict__ W1, const float* __restrict__ b1,
                               const float* __restrict__ W2, const float* __restrict__ b2,
                               const int* __restrict__ counts, const int* __restrict__ tok,
                               const float* __restrict__ wgt,
                               float* __restrict__ out, int T, int maxTiles) {
  extern __shared__ float smem[];
  float* Xs  = smem;                              // M_TILE * SX
  float* As  = Xs + M_TILE * SX;                  // M_TILE * SA
  int*   tks = (int*)(As + M_TILE * SA);          // M_TILE
  float* wts = (float*)(tks + M_TILE);            // M_TILE

  int e    = blockIdx.x / maxTiles;
  int tile = blockIdx.x % maxTiles;
  if (tile * M_TILE >= counts[e]) return;          // uniform per block: EXEC stays full

  int tid  = threadIdx.x;
  int wave = tid >> 5;
  int lane = tid & 31;
  int half = lane >> 4;
  int ln16 = lane & 15;

  if (tid < M_TILE) {
    int row = tile * M_TILE + tid;
    if (row < T) { tks[tid] = tok[e * T + row]; wts[tid] = wgt[e * T + row]; }
    else         { tks[tid] = 0;               wts[tid] = 0.f; }
  }
  __syncthreads();

  { // gather X rows via async global->LDS b128 (8 threads per row)
    int r  = tid >> 3;
    int c8 = tid & 7;
    const float* src = x + (size_t)tks[r] * DIM;
    float* dst = Xs + r * SX;
    for (int c = c8 * 4; c < DIM; c += 32)
      async_copy_b128(dst + c, src + c);
    async_wait_all();
  }
  __syncthreads();

  const float* W1e = W1 + (size_t)e * DIM * HCOLS;
  const float* W2e = W2 + (size_t)e * FF * DIM;
  const float* b1e = b1 + e * HCOLS;
  const float* b2e = b2 + e * DIM;

  const v8f vzero = {0.f, 0.f, 0.f, 0.f, 0.f, 0.f, 0.f, 0.f};
  v8f cy[2][8];
#pragma unroll
  for (int mt = 0; mt < 2; ++mt)
#pragma unroll
    for (int j = 0; j < 8; ++j) cy[mt][j] = vzero;

  for (int ch = 0; ch < FF / F_CHUNK; ++ch) {
    int ca_base = ch * F_CHUNK + wave * 16;        // this wave's 16 a-columns
    // ---- stage 1: K = 1024, compute a-half and g-half tiles together ----
    v8f ca0 = vzero, ca1 = vzero, cg0 = vzero, cg1 = vzero;
    const float* pA = W1e + ca_base + ln16;
    const float* pG = W1e + (FF + ca_base) + ln16;
    for (int k = 0; k < DIM / 4; ++k) {
      int kb = 4 * k + 2 * half;
      v2f a0 = *(const v2f*)(Xs + ln16 * SX + kb);
      v2f a1 = *(const v2f*)(Xs + (16 + ln16) * SX + kb);
      v2f bA; bA.x = pA[(size_t)kb * HCOLS]; bA.y = pA[(size_t)(kb + 1) * HCOLS];
      v2f bG; bG.x = pG[(size_t)kb * HCOLS]; bG.y = pG[(size_t)(kb + 1) * HCOLS];
      ca0 = wmma4(a0, bA, ca0);
      ca1 = wmma4(a1, bA, ca1);
      cg0 = wmma4(a0, bG, cg0);
      cg1 = wmma4(a1, bG, cg1);
    }
    // ---- bias + SwiGLU (exact gelu), write act chunk to LDS ----
    float b1a = b1e[ca_base + ln16];
    float b1g = b1e[FF + ca_base + ln16];
#pragma unroll
    for (int r = 0; r < 8; ++r) {
      float a0v = ca0[r] + b1a, g0v = cg0[r] + b1g;
      float a1v = ca1[r] + b1a, g1v = cg1[r] + b1g;
      float act0 = a0v * (0.5f * g0v * (1.f + erff(g0v * 0.70710678118654752f)));
      float act1 = a1v * (0.5f * g1v * (1.f + erff(g1v * 0.70710678118654752f)));
      int rowm = r + 8 * half;
      As[rowm * SA + wave * 16 + ln16]        = act0;
      As[(16 + rowm) * SA + wave * 16 + ln16] = act1;
    }
    __syncthreads();
    // ---- stage 2: y[32, wave's 128 cols] += act_chunk @ W2 ----
    const float* pW2 = W2e + (size_t)(ch * F_CHUNK) * DIM + wave * 128 + ln16;
    for (int k = 0; k < F_CHUNK / 4; ++k) {
      int kb = 4 * k + 2 * half;
      v2f a0 = *(const v2f*)(As + ln16 * SA + kb);
      v2f a1 = *(const v2f*)(As + (16 + ln16) * SA + kb);
#pragma unroll
      for (int jn = 0; jn < 8; ++jn) {
        v2f b;
        b.x = pW2[(size_t)kb * DIM + jn * 16];
        b.y = pW2[(size_t)(kb + 1) * DIM + jn * 16];
        cy[0][jn] = wmma4(a0, b, cy[0][jn]);
        cy[1][jn] = wmma4(a1, b, cy[1][jn]);
      }
    }
    __syncthreads();
  }

  // ---- epilogue: out[token] += w * (y + b2), f32 hw atomics ----
#pragma unroll
  for (int mt = 0; mt < 2; ++mt) {
#pragma unroll
    for (int r = 0; r < 8; ++r) {
      int rl = mt * 16 + 8 * half + r;
      int token = tks[rl];
      float wt  = wts[rl];
      float* orow = out + (size_t)token * DIM;
#pragma unroll
      for (int jn = 0; jn < 8; ++jn) {
        int col = wave * 128 + jn * 16 + ln16;
        float v = (cy[mt][jn][r] + b2e[col]) * wt;
        unsafeAtomicAdd(orow + col, v);
      }
    }
  }
}

extern "C" void kernel_launch(void* const* d_in, const int* in_sizes, int n_in,
                              void* d_out, int out_size, void* d_ws, size_t ws_size,
                              hipStream_t stream) {
  const float* x  = (const float*)d_in[0];
  const float* Wg = (const float*)d_in[1];
  const float* bg = (const float*)d_in[2];
  const float* W1 = (const float*)d_in[3];
  const float* b1 = (const float*)d_in[4];
  const float* W2 = (const float*)d_in[5];
  const float* b2 = (const float*)d_in[6];
  float* out = (float*)d_out;

  int T = in_sizes[0] / DIM;                 // 8192
  int maxTiles = (T + M_TILE - 1) / M_TILE;  // 256

  // workspace layout
  int*   counts  = (int*)d_ws;
  int*   toklist = (int*)((char*)d_ws + 256);
  float* wlist   = (float*)((char*)d_ws + 256 + (size_t)NEXP * T * sizeof(int));

  int n4 = out_size / 4;
  zero_out_kernel<<<(n4 + 255) / 256, 256, 0, stream>>>((float4*)out, n4);

  int nl = NEXP * T;
  init_lists_kernel<<<(nl + 255) / 256, 256, 0, stream>>>(counts, toklist, wlist, nl);

  router_kernel<<<(T + 7) / 8, 256, 0, stream>>>(x, Wg, bg, counts, toklist, wlist, T);

  size_t smem = (size_t)(M_TILE * SX + M_TILE * SA) * sizeof(float) + M_TILE * 8;
  hipFuncSetAttribute((const void*)moe_mlp_kernel,
                      hipFuncAttributeMaxDynamicSharedMemorySize, (int)smem);
  moe_mlp_kernel<<<dim3(NEXP * maxTiles), 256, smem, stream>>>(
      x, W1, b1, W2, b2, counts, toklist, wlist, out, T, maxTiles);
}